// TextEncoder_60378650247808
// MI455X (gfx1250) — compile-verified
//
#include <hip/hip_runtime.h>
#include <hip/hip_bf16.h>
#include <math.h>

// Problem constants (match reference)
#define NSEQ      64
#define OUT_DIM   480                 // NT*SEM = 30*16
#define WDIM      512
#define NTILES_N  30                  // 480/16
#define NTILES_M  256                 // 4096/16
#define NSPANS    2016                // 64*63/2
#define PREF_ROWS 65                  // N+1
#define PREF_STRIDE (PREF_ROWS * OUT_DIM)   // floats per batch in prefix buffer

typedef __attribute__((ext_vector_type(2))) float v2f;
typedef __attribute__((ext_vector_type(8))) float v8f;

// ---------------------------------------------------------------------------
// Kernel 1: gather(emb[x]) @ w_out^T + b_out, masked, written into prefix[:,1:,:]
// One wave32 per 16x16 output tile, f32 WMMA over K=512 in steps of 4.
// ---------------------------------------------------------------------------
__global__ __launch_bounds__(256) void te_gemm_kernel(
    const int* __restrict__ x, const int* __restrict__ lengths,
    const float* __restrict__ emb, const float* __restrict__ w,
    const float* __restrict__ bias, float* __restrict__ prefix)
{
    const int lane    = threadIdx.x & 31;
    const int wave    = threadIdx.x >> 5;
    const int tile    = blockIdx.x * 8 + wave;       // 0..7679 exactly
    const int mtile   = tile / NTILES_N;             // 0..255
    const int ntile   = tile % NTILES_N;             // 0..29

    const int half    = lane >> 4;                   // 0 or 1
    const int lid     = lane & 15;

    // A: row of gathered embedding (M = lid); per k-step lane loads K = k+2*half .. +1
    const int row     = mtile * 16 + lid;            // 0..4095
    const int wid     = x[row];
    const float* arow = emb + (size_t)wid * WDIM;
    // B: w_out row (N = lid); same K offsets -> contiguous float2
    const float* brow = w + (size_t)(ntile * 16 + lid) * WDIM;

    v8f acc = {};
    #pragma unroll 8
    for (int k = 0; k < WDIM; k += 4) {
        const int kk = k + 2 * half;
        v2f a = *(const v2f*)(arow + kk);
        v2f b = *(const v2f*)(brow + kk);
        acc = __builtin_amdgcn_wmma_f32_16x16x4_f32(
                  /*neg_a=*/false, a, /*neg_b=*/false, b,
                  /*c_mod=*/(short)0, acc, /*reuse_a=*/false, /*reuse_b=*/false);
    }

    // D layout: lane L, reg v -> D[M = v + 8*half, N = lid]
    const int o     = ntile * 16 + lid;
    const float bo  = bias[o];
    const int bidx  = mtile >> 2;                    // 4 M-tiles per batch, no straddle
    const int len_b = lengths[bidx];
    const int nbase = (mtile & 3) * 16;
    float* pb = prefix + (size_t)bidx * PREF_STRIDE + o;
    #pragma unroll
    for (int v = 0; v < 8; ++v) {
        const int m = v + 8 * half;
        const int n = nbase + m;                     // position within sequence
        float val = acc[v] + bo;
        if (n >= len_b) val = 0.0f;
        pb[(size_t)(n + 1) * OUT_DIM] = val;         // rows 1..64; row 0 zeroed in cumsum
    }
}

// ---------------------------------------------------------------------------
// Kernel 2: in-place inclusive scan along sequence; prefix[b][0][o] = 0.
// One thread per (b, o): 64*480 = 30720 independent scans, coalesced in o.
// ---------------------------------------------------------------------------
__global__ __launch_bounds__(256) void te_cumsum_kernel(float* __restrict__ prefix)
{
    const int t = blockIdx.x * 256 + threadIdx.x;
    if (t >= NSEQ * OUT_DIM) return;
    const int b = t / OUT_DIM;
    const int o = t % OUT_DIM;
    float* p = prefix + (size_t)b * PREF_STRIDE + o;
    p[0] = 0.0f;
    float run = 0.0f;
    #pragma unroll 4
    for (int n = 1; n <= NSEQ; ++n) {
        run += p[(size_t)n * OUT_DIM];
        p[(size_t)n * OUT_DIM] = run;
    }
}

// ---------------------------------------------------------------------------
// Kernel 3: span features + L2-normalize over SEM=16.
// One thread per (b, s, nt): 4x 128-bit loads per prefix row (L2-hot),
// 64 B contiguous store per thread -> fully coalesced 2 KB per wave.
// ---------------------------------------------------------------------------
__global__ __launch_bounds__(256) void te_span_kernel(
    const float* __restrict__ prefix, float* __restrict__ out)
{
    const int total = 64 * NSPANS * NTILES_N;        // 3,870,720
    const int t = blockIdx.x * 256 + threadIdx.x;
    if (t >= total) return;

    const int nt  = t % NTILES_N;
    const int rem = t / NTILES_N;
    const int s   = rem % NSPANS;
    const int b   = rem / NSPANS;

    // Decode span index: spans ordered by k=1..63, within k by l=0..63-k.
    // f(k) = 64k - k(k+1)/2 spans have length-index <= k. Invert + fixup.
    int kk = (int)floorf((127.0f - sqrtf((float)(16129 - 8 * s))) * 0.5f) + 1;
    if (kk < 1)  kk = 1;
    if (kk > 63) kk = 63;
    while (64 * kk - (kk * (kk + 1)) / 2 <= s) ++kk;
    while (64 * (kk - 1) - ((kk - 1) * kk) / 2 > s) --kk;
    const int base = 64 * (kk - 1) - ((kk - 1) * kk) / 2;
    const int l = s - base;
    const int r = l + kk;                            // rs; read prefix row r+1

    const float* pl = prefix + (size_t)b * PREF_STRIDE + (size_t)l * OUT_DIM + nt * 16;
    const float* pr = prefix + (size_t)b * PREF_STRIDE + (size_t)(r + 1) * OUT_DIM + nt * 16;

    float4 d0, d1, d2, d3;
    {
        const float4* pl4 = (const float4*)pl;
        const float4* pr4 = (const float4*)pr;
        float4 a0 = pr4[0], a1 = pr4[1], a2 = pr4[2], a3 = pr4[3];
        float4 b0 = pl4[0], b1 = pl4[1], b2 = pl4[2], b3 = pl4[3];
        d0 = make_float4(a0.x - b0.x, a0.y - b0.y, a0.z - b0.z, a0.w - b0.w);
        d1 = make_float4(a1.x - b1.x, a1.y - b1.y, a1.z - b1.z, a1.w - b1.w);
        d2 = make_float4(a2.x - b2.x, a2.y - b2.y, a2.z - b2.z, a2.w - b2.w);
        d3 = make_float4(a3.x - b3.x, a3.y - b3.y, a3.z - b3.z, a3.w - b3.w);
    }
    float ss = d0.x*d0.x + d0.y*d0.y + d0.z*d0.z + d0.w*d0.w
             + d1.x*d1.x + d1.y*d1.y + d1.z*d1.z + d1.w*d1.w
             + d2.x*d2.x + d2.y*d2.y + d2.z*d2.z + d2.w*d2.w
             + d3.x*d3.x + d3.y*d3.y + d3.z*d3.z + d3.w*d3.w;
    const float inv = rsqrtf(ss);

    float4* o4 = (float4*)(out + (size_t)t * 16);
    o4[0] = make_float4(d0.x*inv, d0.y*inv, d0.z*inv, d0.w*inv);
    o4[1] = make_float4(d1.x*inv, d1.y*inv, d1.z*inv, d1.w*inv);
    o4[2] = make_float4(d2.x*inv, d2.y*inv, d2.z*inv, d2.w*inv);
    o4[3] = make_float4(d3.x*inv, d3.y*inv, d3.z*inv, d3.w*inv);
}

// ---------------------------------------------------------------------------
extern "C" void kernel_launch(void* const* d_in, const int* in_sizes, int n_in,
                              void* d_out, int out_size, void* d_ws, size_t ws_size,
                              hipStream_t stream)
{
    const int*   x       = (const int*)d_in[0];      // [64*64]
    const int*   lengths = (const int*)d_in[1];      // [64]
    const float* emb     = (const float*)d_in[2];    // [10000*512]
    const float* w_out   = (const float*)d_in[3];    // [480*512]
    const float* b_out   = (const float*)d_in[4];    // [480]
    float*       out     = (float*)d_out;            // [64*2016*30*16]
    float*       prefix  = (float*)d_ws;             // [64*65*480] = ~8 MB

    // K1: 7680 tiles, 8 waves (256 thr) per block
    te_gemm_kernel<<<960, 256, 0, stream>>>(x, lengths, emb, w_out, b_out, prefix);

    // K2: 30720 scans
    te_cumsum_kernel<<<(NSEQ * OUT_DIM + 255) / 256, 256, 0, stream>>>(prefix);

    // K3: 3,870,720 threads, one per (b, span, nt)
    const int total = 64 * NSPANS * NTILES_N;
    te_span_kernel<<<(total + 255) / 256, 256, 0, stream>>>(prefix, out);
}